// TimePointMasker_90263032692750
// MI455X (gfx1250) — compile-verified
//
#include <hip/hip_runtime.h>
#include <stdint.h>

// ---------------------------------------------------------------------------
// TimePointMasker for MI455X (gfx1250, wave32).
//
// Mask is independent of x's values -> never read d_in (saves 512 MiB of HBM
// read traffic; kernel is pure-store bound: 512 MiB @ 23.3 TB/s ~ 23 us).
// JAX threefry2x32 pairs flat counter i with i + N/2; since N/2 is a whole
// number of rows (16384), one workgroup produces keys for rows j and j+16384
// from the same threefry calls (x0 lane / x1 lane), halving PRNG compute.
//
// Exact selection of the 614 smallest (key, idx) per row:
//   * 23-bit ordering key = random_bits >> 9 (monotone with JAX uniforms)
//   * 256-bin histogram of top 8 bits (LDS atomics)
//   * 16x16 within-row inclusive prefix via 4 chained V_WMMA_F32_16X16X4_F32
//   * pivot bin refined with (low15 << 12 | idx) composites (stable tie-break,
//     identical to stable argsort rank semantics)
// ---------------------------------------------------------------------------

typedef __attribute__((ext_vector_type(2))) float v2f;
typedef __attribute__((ext_vector_type(8))) float v8f;

#define T_LEN      4096
#define NUM_MASK   614          // int(0.15 * 4096)
#define HALF_ROWS  16384        // (64*512)/2
#define HALF_ELEMS 67108864u    // 64*512*4096/2

__device__ __forceinline__ uint32_t rotl32(uint32_t x, int n) {
  return (x << n) | (x >> (32 - n));
}

// threefry2x32 with key = jax.random.key(42) -> (0, 42), 20 rounds.
__device__ __forceinline__ void threefry2x32_seed42(uint32_t x0, uint32_t x1,
                                                    uint32_t& o0, uint32_t& o1) {
  const uint32_t ks0 = 0u, ks1 = 42u;
  const uint32_t ks2 = ks0 ^ ks1 ^ 0x1BD11BDAu;
  x0 += ks0; x1 += ks1;
#define TF_ROUND(r) { x0 += x1; x1 = rotl32(x1, (r)); x1 ^= x0; }
  TF_ROUND(13) TF_ROUND(15) TF_ROUND(26) TF_ROUND(6)
  x0 += ks1; x1 += ks2 + 1u;
  TF_ROUND(17) TF_ROUND(29) TF_ROUND(16) TF_ROUND(24)
  x0 += ks2; x1 += ks0 + 2u;
  TF_ROUND(13) TF_ROUND(15) TF_ROUND(26) TF_ROUND(6)
  x0 += ks0; x1 += ks1 + 3u;
  TF_ROUND(17) TF_ROUND(29) TF_ROUND(16) TF_ROUND(24)
  x0 += ks1; x1 += ks2 + 4u;
  TF_ROUND(13) TF_ROUND(15) TF_ROUND(26) TF_ROUND(6)
  x0 += ks2; x1 += ks0 + 5u;
#undef TF_ROUND
  o0 = x0; o1 = x1;
}

__global__ __launch_bounds__(256)
void TimePointMasker_kernel(float* __restrict__ out) {
  __shared__ uint32_t keysA[T_LEN];     // 23-bit keys, row j
  __shared__ uint32_t keysB[T_LEN];     // 23-bit keys, row j + HALF_ROWS
  __shared__ uint32_t hist[256];        // histogram of key >> 15
  __shared__ float    prefixLDS[256];   // within-row inclusive prefix (WMMA out)
  __shared__ float    rowOff[16];       // exclusive row offsets
  __shared__ uint32_t list_[T_LEN];     // pivot-bin composites
  __shared__ int      listCount;
  __shared__ int      selBin;
  __shared__ int      selQuota;

  const int      tid = threadIdx.x;     // 0..255 (8 waves of 32)
  const uint32_t row = blockIdx.x;      // 0..HALF_ROWS-1

  // ---- phase 0: PRNG keys for both paired rows -------------------------
  for (int k = 0; k < 16; ++k) {
    const uint32_t t = (uint32_t)(k * 256 + tid);
    const uint32_t p = row * (uint32_t)T_LEN + t;   // pair counter < HALF_ELEMS
    uint32_t o0, o1;
    threefry2x32_seed42(p, p + HALF_ELEMS, o0, o1);
    keysA[t] = o0 >> 9;   // ordering-equivalent to jax uniform floats
    keysB[t] = o1 >> 9;
  }

  for (int sel = 0; sel < 2; ++sel) {
    const uint32_t* keys = sel ? keysB : keysA;
    float* out_row = out + (size_t)(row + (uint32_t)sel * HALF_ROWS) * T_LEN;

    // ---- reset shared state ------------------------------------------
    hist[tid] = 0u;
    if (tid == 0) listCount = 0;
    __syncthreads();

    // ---- phase 1: 256-bin histogram of top 8 key bits ----------------
    for (int k = 0; k < 16; ++k) {
      const int t = k * 256 + tid;              // same t this thread wrote
      atomicAdd(&hist[keys[t] >> 15], 1u);
    }
    __syncthreads();

    // ---- phase 2: within-row inclusive prefix via WMMA ---------------
    // RowPrefix[i][j] = sum_{k<=j} H[i][k], H[i][k] = hist[16*i+k].
    // K=16 built from 4 chained 16x16x4 f32 WMMAs:
    //   A slice g = H[:, 4g:4g+4], B slice g = U[4g:4g+4, :], U[k][j]=(k<=j).
    // A layout (ISA 7.12.2): lane<16 -> M=lane, K={4g+0,4g+1}; lane>=16 ->
    // M=lane-16, K={4g+2,4g+3}.  B mirrors K mapping with N=lane%16.
    if (tid < 32) {
      const int m  = tid & 15;     // A row / B column / D column
      const int hi = tid >> 4;     // 0: K pair (0,1); 1: K pair (2,3)
      v8f acc = {};
      for (int g = 0; g < 4; ++g) {
        const int k0 = 4 * g + 2 * hi;
        const int k1 = k0 + 1;
        v2f a, b;
        a.x = (float)hist[m * 16 + k0];
        a.y = (float)hist[m * 16 + k1];
        b.x = (k0 <= m) ? 1.0f : 0.0f;
        b.y = (k1 <= m) ? 1.0f : 0.0f;
        acc = __builtin_amdgcn_wmma_f32_16x16x4_f32(
            /*neg_a=*/false, a, /*neg_b=*/false, b,
            /*c_mod=*/(short)0, acc, /*reuse_a=*/false, /*reuse_b=*/false);
      }
      // D layout: VGPR v -> M = v + 8*hi, N = m
#pragma unroll
      for (int v = 0; v < 8; ++v)
        prefixLDS[(v + 8 * hi) * 16 + m] = acc[v];
    }
    __syncthreads();

    // ---- phase 3: row offsets (16-step scan) + parallel pivot find ---
    if (tid == 0) {
      float cum = 0.0f;
      for (int i = 0; i < 16; ++i) { rowOff[i] = cum; cum += prefixLDS[i * 16 + 15]; }
    }
    __syncthreads();
    {
      const int b    = tid;                                  // one bin per thread
      const int incl = (int)(rowOff[b >> 4] + prefixLDS[b] + 0.5f);
      const int excl = incl - (int)hist[b];
      if (excl < NUM_MASK && incl >= NUM_MASK) {             // unique crossing
        selBin   = b;
        selQuota = NUM_MASK - excl;
      }
    }
    __syncthreads();
    const int bsel  = selBin;
    const int quota = selQuota;

    // ---- phase 4: collect pivot-bin elements -------------------------
    for (int k = 0; k < 16; ++k) {
      const int      t   = k * 256 + tid;
      const uint32_t key = keys[t];
      if ((int)(key >> 15) == bsel) {
        const int pos = atomicAdd(&listCount, 1);
        list_[pos] = ((key & 0x7FFFu) << 12) | (uint32_t)t;  // (low15, idx)
      }
    }
    __syncthreads();
    const int mcnt = listCount;

    // ---- phase 5: emit mask, coalesced float4 stores -----------------
    for (int k = 0; k < 4; ++k) {
      const int t0 = k * 1024 + tid * 4;
      float4 v4;
      float* vp = &v4.x;
#pragma unroll
      for (int j = 0; j < 4; ++j) {
        const int      t   = t0 + j;
        const uint32_t key = keys[t];
        const int      bin = (int)(key >> 15);
        float v;
        if (bin < bsel)      v = 0.0f;
        else if (bin > bsel) v = 1.0f;
        else {
          const uint32_t comp = ((key & 0x7FFFu) << 12) | (uint32_t)t;
          int rank = 0;
          for (int i = 0; i < mcnt; ++i) rank += (list_[i] < comp) ? 1 : 0;
          v = (rank < quota) ? 0.0f : 1.0f;
        }
        vp[j] = v;
      }
      *reinterpret_cast<float4*>(out_row + t0) = v4;
    }
    __syncthreads();   // shared structures reused by next sel iteration
  }
}

extern "C" void kernel_launch(void* const* d_in, const int* in_sizes, int n_in,
                              void* d_out, int out_size, void* d_ws, size_t ws_size,
                              hipStream_t stream) {
  (void)d_in; (void)in_sizes; (void)n_in; (void)d_ws; (void)ws_size; (void)out_size;
  float* out = (float*)d_out;
  TimePointMasker_kernel<<<dim3(HALF_ROWS), dim3(256), 0, stream>>>(out);
}